// GenuineTrainedDistillationTransformer_137438953637
// MI455X (gfx1250) — compile-verified
//
#include <hip/hip_runtime.h>
#include <hip/hip_bf16.h>

typedef float v2f __attribute__((ext_vector_type(2)));
typedef float v8f __attribute__((ext_vector_type(8)));

#define EMBED   1024
#define LATD    24
#define CODED   12
#define TILE_M  16
#define THREADS 256
#define NBLOCKS 512

// ---- LDS partition (float offsets) ----
#define OFF_B1    0        // W_el B-fragments: [256 ksteps][32 n][4 k4] = 32768 floats (128KB)
#define OFF_B3    32768    // W_le B-fragments: [64 nt][6 s][16 n][4 k4] = 24576 floats (96KB)
#define OFF_GOL   57344    // golay [12][24] = 288
#define OFF_BEL   57632    // b_el 24
#define OFF_BLE   57656    // b_le 1024
#define OFF_LAT   58680    // lat accum [16][32] = 512
#define OFF_SUMSQ 59192    // 16
#define OFF_CORR  59208    // corr [16][24] = 384
#define OFF_INE2  59592    // 16
#define OFF_RNORM 59608    // 16
#define OFF_SC2   59624    // 16
#define SMEM_FLOATS 59640

extern "C" __global__ __launch_bounds__(THREADS)
void golay_lattice_fused(const float* __restrict__ data,
                         const float* __restrict__ W_el,
                         const float* __restrict__ b_el,
                         const float* __restrict__ W_le,
                         const float* __restrict__ b_le,
                         const float* __restrict__ golay,
                         const float* __restrict__ ecs_p,
                         const float* __restrict__ ep_p,
                         float* __restrict__ out,
                         int ntiles)
{
    extern __shared__ __align__(16) float smem[];
    const int tid  = threadIdx.x;
    const int lane = tid & 31;
    const int wave = tid >> 5;
    const int m16  = lane & 15;     // N (or M-row group) index within fragment
    const int hi   = lane >> 4;     // lane half: selects K pair {0,1} vs {2,3}
    const int kb   = hi << 1;       // K offset 0 or 2
    const float ecs = ecs_p[0];
    const float ep  = ep_p[0];
    const float inv_ecs = 1.0f / ecs;

    // ---------------- one-time LDS init ----------------
    // W_el -> B1 fragments: B1[s][n][k4] = (n<24) ? W_el[n][4s+k4] : 0
    for (int i = tid; i < 32768; i += THREADS) {
        int s = i >> 7, n = (i >> 2) & 31, k4 = i & 3;
        smem[OFF_B1 + i] = (n < LATD) ? W_el[n * EMBED + (s << 2) + k4] : 0.0f;
    }
    // W_le -> B3 fragments: B3[nt][s][n][k4] = W_le[nt*16+n][4s+k4]   (4s+k4 < 24 always)
    for (int i = tid; i < 24576; i += THREADS) {
        int k4 = i & 3, n = (i >> 2) & 15, t = i >> 6;
        int s = t % 6, nt = t / 6;
        smem[OFF_B3 + i] = W_le[(nt * 16 + n) * LATD + (s << 2) + k4];
    }
    for (int i = tid; i < 288;  i += THREADS) smem[OFF_GOL + i] = golay[i];
    for (int i = tid; i < LATD; i += THREADS) smem[OFF_BEL + i] = b_el[i];
    for (int i = tid; i < EMBED; i += THREADS) smem[OFF_BLE + i] = b_le[i];
    __syncthreads();

    for (int tile = blockIdx.x; tile < ntiles; tile += gridDim.x) {
        const int tok0 = tile * TILE_M;

        // zero per-tile reduction buffers
        for (int i = tid; i < 512; i += THREADS) smem[OFF_LAT + i] = 0.0f;
        if (tid < 16) { smem[OFF_SUMSQ + tid] = 0.0f; smem[OFF_RNORM + tid] = 0.0f; }
        __syncthreads();

        // ---------------- phase 1: lat = data @ W_el^T  (M=16, N=24->2x16, K=1024/8 waves)
        v8f acc0 = {0,0,0,0,0,0,0,0};
        v8f acc1 = {0,0,0,0,0,0,0,0};
        float ssq = 0.0f;
        const float* arow = data + (size_t)(tok0 + m16) * EMBED + kb;
        const int s0 = wave * 32;
        #pragma unroll 1
        for (int g = 0; g < 4; ++g) {
            // one prefetch per 8 K-steps, WGP scope (pull into all levels)
            if (g != 3)
                __builtin_prefetch(arow + ((s0 + (g << 3) + 8) << 2), 0, 3);
            #pragma unroll
            for (int t = 0; t < 8; ++t) {
                const int sg = s0 + (g << 3) + t;
                v2f A = *(const v2f*)(arow + (sg << 2));                // global_load_b64
                ssq = __builtin_fmaf(A.x, A.x, __builtin_fmaf(A.y, A.y, ssq));
                const float* bb = &smem[OFF_B1 + (sg << 7) + kb];
                v2f B0 = *(const v2f*)(bb + (m16 << 2));                // ds_load_2addr_b64
                v2f B1 = *(const v2f*)(bb + ((m16 + 16) << 2));
                acc0 = __builtin_amdgcn_wmma_f32_16x16x4_f32(false, A, false, B0,
                                                             (short)0, acc0, false, false);
                acc1 = __builtin_amdgcn_wmma_f32_16x16x4_f32(false, A, false, B1,
                                                             (short)0, acc1, false, false);
            }
        }
        // ||data||^2 partials: lane l and l^16 cover K pairs {0,1}/{2,3} of token m16
        ssq += __shfl_xor(ssq, 16, 32);
        if (hi == 0) atomicAdd(&smem[OFF_SUMSQ + m16], ssq);        // ds_add_f32
        // cross-wave accumulate lat tile
        #pragma unroll
        for (int r = 0; r < 8; ++r) {
            const int mm = r + (hi << 3);
            atomicAdd(&smem[OFF_LAT + mm * 32 + m16],      acc0[r]);
            atomicAdd(&smem[OFF_LAT + mm * 32 + 16 + m16], acc1[r]);
        }
        __syncthreads();

        // ---------------- phase 2: golay encode/quantize/renorm/decode (tiny, wave 0)
        if (wave == 0 && hi == 0) {
            const int m = m16;
            float latv[LATD];
            #pragma unroll
            for (int l = 0; l < LATD; ++l)
                latv[l] = smem[OFF_LAT + m * 32 + l] + smem[OFF_BEL + l];
            float enc[CODED]; float np2 = 0.0f;
            #pragma unroll
            for (int c = 0; c < CODED; ++c) {
                float g = 0.0f;
                #pragma unroll
                for (int l = 0; l < LATD; ++l)
                    g = __builtin_fmaf(latv[l], smem[OFF_GOL + c * LATD + l], g);
                float p = rintf(g * inv_ecs) * ecs;   // RNE == jnp.round
                enc[c] = p;
                np2 = __builtin_fmaf(p, p, np2);
            }
            const float in_e = sqrtf(smem[OFF_SUMSQ + m]);
            const float sc1  = in_e / (sqrtf(np2) + 1e-8f) * ep;
            float e2 = 0.0f;
            #pragma unroll
            for (int c = 0; c < CODED; ++c) {
                enc[c] *= sc1;
                e2 = __builtin_fmaf(enc[c], enc[c], e2);
            }
            smem[OFF_INE2 + m] = sqrtf(e2);
            #pragma unroll
            for (int l = 0; l < LATD; ++l) {
                float d = 0.0f;
                #pragma unroll
                for (int c = 0; c < CODED; ++c)
                    d = __builtin_fmaf(enc[c], smem[OFF_GOL + c * LATD + l], d);
                smem[OFF_CORR + m * LATD + l] = (fabsf(d) > ecs) ? d : 0.0f;
            }
        }
        __syncthreads();

        // ---------------- phase 3: res = corr @ W_le^T  (M=16, K=24, N=1024/8 waves)
        v8f racc[8];
        float ss[8];
        #pragma unroll
        for (int r = 0; r < 8; ++r) ss[r] = 0.0f;
        #pragma unroll
        for (int j = 0; j < 8; ++j) {
            const int nt = (wave << 3) + j;
            v8f acc = {0,0,0,0,0,0,0,0};
            #pragma unroll
            for (int s = 0; s < 6; ++s) {
                v2f A = *(const v2f*)&smem[OFF_CORR + m16 * LATD + (s << 2) + kb];
                v2f B = *(const v2f*)&smem[OFF_B3 + ((nt * 6 + s) << 6) + (m16 << 2) + kb];
                acc = __builtin_amdgcn_wmma_f32_16x16x4_f32(false, A, false, B,
                                                            (short)0, acc, false, false);
            }
            const float bl = smem[OFF_BLE + (nt << 4) + m16];
            #pragma unroll
            for (int r = 0; r < 8; ++r) {
                float v = acc[r] + bl;
                acc[r] = v;
                ss[r] = __builtin_fmaf(v, v, ss[r]);
            }
            racc[j] = acc;
        }
        // ||res||^2: reduce across the 16 lanes of each half, token = r + hi*8
        #pragma unroll
        for (int r = 0; r < 8; ++r) {
            float v = ss[r];
            v += __shfl_xor(v, 1, 32);
            v += __shfl_xor(v, 2, 32);
            v += __shfl_xor(v, 4, 32);
            v += __shfl_xor(v, 8, 32);
            if (m16 == 0) atomicAdd(&smem[OFF_RNORM + r + (hi << 3)], v);
        }
        __syncthreads();
        if (wave == 0 && hi == 0)
            smem[OFF_SC2 + m16] = smem[OFF_INE2 + m16] /
                                  (sqrtf(smem[OFF_RNORM + m16]) + 1e-8f) * ep;
        __syncthreads();

        float s2[8];
        #pragma unroll
        for (int r = 0; r < 8; ++r) s2[r] = smem[OFF_SC2 + r + (hi << 3)];
        #pragma unroll
        for (int j = 0; j < 8; ++j) {
            const int nt = (wave << 3) + j;
            float* orow = out + (size_t)tok0 * EMBED + (nt << 4) + m16;
            #pragma unroll
            for (int r = 0; r < 8; ++r)
                orow[(size_t)(r + (hi << 3)) * EMBED] = racc[j][r] * s2[r];
        }
        // next-iteration zeroing only touches LAT/SUMSQ/RNORM and is fenced by
        // the barrier after zeroing, so no trailing barrier needed here.
    }
}

extern "C" void kernel_launch(void* const* d_in, const int* in_sizes, int n_in,
                              void* d_out, int out_size, void* d_ws, size_t ws_size,
                              hipStream_t stream) {
    const float* data  = (const float*)d_in[0];
    const float* W_el  = (const float*)d_in[1];
    const float* b_el  = (const float*)d_in[2];
    const float* W_le  = (const float*)d_in[3];
    const float* b_le  = (const float*)d_in[4];
    const float* golay = (const float*)d_in[5];
    const float* ecs   = (const float*)d_in[6];
    const float* ep    = (const float*)d_in[7];
    float* out = (float*)d_out;

    const int ntok   = in_sizes[0] / EMBED;     // B*T = 32768
    const int ntiles = ntok / TILE_M;           // 2048
    const size_t smem_bytes = (size_t)SMEM_FLOATS * sizeof(float);  // ~233 KB

    hipLaunchKernelGGL(golay_lattice_fused,
                       dim3(NBLOCKS), dim3(THREADS), smem_bytes, stream,
                       data, W_el, b_el, W_le, b_le, golay, ecs, ep, out, ntiles);
}